// IrrepsToHessian_42356967473304
// MI455X (gfx1250) — compile-verified
//
#include <hip/hip_runtime.h>

typedef float v2f __attribute__((ext_vector_type(2)));
typedef float v8f __attribute__((ext_vector_type(8)));

#define MDIM 32

// CART_B with the per-m S/A coefficients pre-folded:
//   rows 0..3 : I/sqrt3, eps[p]/sqrt2 (unchanged)
//   rows 4..7 : S[m] * (1/sqrt2)  -> 0.5 entries
//   row  8    : S[4] * (1/sqrt6)  -> 1/6, 1/6, -1/3
__device__ __constant__ float c_BM[81] = {
  0.5773502691896258f,0,0, 0,0.5773502691896258f,0, 0,0,0.5773502691896258f,
  0,0,0, 0,0,0.7071067811865476f, 0,-0.7071067811865476f,0,
  0,0,-0.7071067811865476f, 0,0,0, 0.7071067811865476f,0,0,
  0,0.7071067811865476f,0, -0.7071067811865476f,0,0, 0,0,0,
  0,0.5f,0, 0.5f,0,0, 0,0,0,
  0,0,0, 0,0,0.5f, 0,0.5f,0,
  0,0,0.5f, 0,0,0, 0.5f,0,0,
  0.5f,0,0, 0,-0.5f,0, 0,0,0,
  0.16666666666666666f,0,0, 0,0.16666666666666666f,0, 0,0,-0.3333333333333333f
};

// ---- Kernel A: per-node weight transforms as exact-FP32 WMMA GEMMs ---------
// s-GEMMs:  (n_nodes x 32) @ (32 x 32)   for w_ss, w_sv
// v-GEMMs:  (3*n_nodes x 32) @ (32 x 32) for w_vv0, w_vs, w_vv1, w_vv2
// One wave per 16x16 output tile, 8x V_WMMA_F32_16X16X4_F32 (K=32).
// All scalar prefactors of the downstream terms are folded into the stores:
//   p_ss *= 1/(M*sqrt2)          p_sv *= 1/(M*3)
//   t0   *= 1/(M*sqrt6)          ts   *= 1/(M*3)
//   t1   *= 1/(M*sqrt18)         t2   *= 1/(M*sqrt5)
__global__ void node_gemm_wmma(const float* __restrict__ nf,
                               const float* __restrict__ w_ss,
                               const float* __restrict__ w_sv,
                               const float* __restrict__ w_vv0,
                               const float* __restrict__ w_vs,
                               const float* __restrict__ w_vv1,
                               const float* __restrict__ w_vv2,
                               float* __restrict__ p_ss,
                               float* __restrict__ p_sv,
                               float* __restrict__ t0,
                               float* __restrict__ ts,
                               float* __restrict__ t1,
                               float* __restrict__ t2,
                               int n_nodes, int rt_s, int rt_v)
{
  int wid    = blockIdx.x * (blockDim.x >> 5) + (threadIdx.x >> 5);
  int sTiles = 4 * rt_s;   // 2 weights * rt_s row-tiles * 2 col-tiles
  int vTiles = 8 * rt_v;   // 4 weights * rt_v row-tiles * 2 col-tiles
  if (wid >= sTiles + vTiles) return;   // wave-uniform: EXEC all-1s for WMMA

  const float SC_PSS = 0.022097086912079612f;  // 1/(32*sqrt2)
  const float SC_PSV = 0.010416666666666666f;  // 1/(32*3)
  const float SC_T0  = 0.012758132117111686f;  // 1/(32*sqrt6)
  const float SC_TS  = 0.010416666666666666f;  // 1/(32*3)
  const float SC_T1  = 0.007365695637359862f;  // 1/(32*sqrt18)
  const float SC_T2  = 0.013975424859373686f;  // 1/(32*sqrt5)

  int lane = threadIdx.x & 31;
  int half = lane >> 4;    // 32-bit A/B frag layout: K = vgpr_idx + 2*half
  int lm   = lane & 15;
  v8f acc = {};

  if (wid < sTiles) {
    int widx = wid / (2 * rt_s);
    int rem  = wid % (2 * rt_s);
    int m0 = (rem >> 1) * 16;
    int n0 = (rem & 1) * 16;
    const float* W  = widx ? w_sv : w_ss;
    float*       D  = widx ? p_sv : p_ss;
    float        sc = widx ? SC_PSV : SC_PSS;
    int row = m0 + lm;
    for (int k0 = 0; k0 < MDIM; k0 += 4) {
      v2f a, b;
#pragma unroll
      for (int p = 0; p < 2; ++p) {
        int u = k0 + half * 2 + p;
        a[p] = (row < n_nodes) ? nf[row * 128 + u] : 0.0f;   // A[m][u] = s[m][u]
        b[p] = W[u * MDIM + n0 + lm];                        // B[u][v] = w[u][v]
      }
      acc = __builtin_amdgcn_wmma_f32_16x16x4_f32(false, a, false, b,
                                                  (short)0, acc, false, false);
    }
#pragma unroll
    for (int g = 0; g < 8; ++g) {        // D: M = g + 8*half, N = lm
      int r = m0 + g + 8 * half;
      if (r < n_nodes) D[r * MDIM + n0 + lm] = sc * acc[g];
    }
  } else {
    int t    = wid - sTiles;
    int widx = t / (2 * rt_v);
    int rem  = t % (2 * rt_v);
    int m0 = (rem >> 1) * 16;
    int n0 = (rem & 1) * 16;
    const float* W  = (widx == 0) ? w_vv0 : (widx == 1) ? w_vs
                    : (widx == 2) ? w_vv1 : w_vv2;
    float*       D  = (widx == 0) ? t0 : (widx == 1) ? ts
                    : (widx == 2) ? t1 : t2;
    float        sc = (widx == 0) ? SC_T0 : (widx == 1) ? SC_TS
                    : (widx == 2) ? SC_T1 : SC_T2;
    int rows = 3 * n_nodes;              // logical row rr = (node, i)
    int rr = m0 + lm;
    int nn = rr / 3, ii = rr % 3;
    for (int k0 = 0; k0 < MDIM; k0 += 4) {
      v2f a, b;
#pragma unroll
      for (int p = 0; p < 2; ++p) {
        int u = k0 + half * 2 + p;
        a[p] = (rr < rows) ? nf[nn * 128 + MDIM + u * 3 + ii] : 0.0f; // v[n][u][i]
        b[p] = W[u * MDIM + n0 + lm];
      }
      acc = __builtin_amdgcn_wmma_f32_16x16x4_f32(false, a, false, b,
                                                  (short)0, acc, false, false);
    }
#pragma unroll
    for (int g = 0; g < 8; ++g) {
      int r = m0 + g + 8 * half;
      if (r < rows) {
        int n2 = r / 3, i2 = r % 3;
        D[n2 * 96 + (n0 + lm) * 3 + i2] = sc * acc[g];   // t[n][v][i]
      }
    }
  }
}

// ---- Kernel B: one wave32 per pair -----------------------------------------
// Lane v owns feature channel v: gathers 14 prescaled row-transform floats
// (coalesced 128B/384B runs) + 4 col floats, computes the per-channel
// contribution to the 9 irrep outputs, butterfly-reduces across the wave,
// then lanes 0..8 fold CART_B and store.
__global__ void pair_contract(const float* __restrict__ nf,
                              const int* __restrict__ layout,
                              const float* __restrict__ p_ss,
                              const float* __restrict__ p_sv,
                              const float* __restrict__ t0,
                              const float* __restrict__ ts,
                              const float* __restrict__ t1,
                              const float* __restrict__ t2,
                              float* __restrict__ out, int n_pairs)
{
  int e = blockIdx.x * (blockDim.x >> 5) + (threadIdx.x >> 5);
  if (e >= n_pairs) return;
  int v = threadIdx.x & 31;
  int r = layout[2 * e];
  int c = layout[2 * e + 1];

  // col features (one node_feats row, fully coalesced)
  const float* fc = nf + (size_t)c * 128;
  float s2  = fc[v];
  float v2x = fc[MDIM + v * 3 + 0];
  float v2y = fc[MDIM + v * 3 + 1];
  float v2z = fc[MDIM + v * 3 + 2];

  // row transforms (prescaled in kernel A)
  size_t r32 = (size_t)r * MDIM, r96 = (size_t)r * 96 + (size_t)v * 3;
  float pss = p_ss[r32 + v];
  float psv = p_sv[r32 + v];
  float t0x = t0[r96 + 0], t0y = t0[r96 + 1], t0z = t0[r96 + 2];
  float tsx = ts[r96 + 0], tsy = ts[r96 + 1], tsz = ts[r96 + 2];
  float t1x = t1[r96 + 0], t1y = t1[r96 + 1], t1z = t1[r96 + 2];
  float t2x = t2[r96 + 0], t2y = t2[r96 + 1], t2z = t2[r96 + 2];

  float y[9];
  // scalar irrep
  y[0] = pss * s2 + t0x * v2x + t0y * v2y + t0z * v2z;
  // vector irrep: s->v, v->s, and cross (eps) terms
  y[1] = psv * v2x + tsx * s2 + (t1y * v2z - t1z * v2y);
  y[2] = psv * v2y + tsy * s2 + (t1z * v2x - t1x * v2z);
  y[3] = psv * v2z + tsz * s2 + (t1x * v2y - t1y * v2x);
  // symmetric-traceless irrep (S-basis coefficients folded into c_BM)
  y[4] = t2x * v2y + t2y * v2x;
  y[5] = t2y * v2z + t2z * v2y;
  y[6] = t2x * v2z + t2z * v2x;
  y[7] = t2x * v2x - t2y * v2y;
  y[8] = t2x * v2x + t2y * v2y - 2.0f * t2z * v2z;

  // wave32 butterfly reduction
#pragma unroll
  for (int off = 16; off >= 1; off >>= 1) {
#pragma unroll
    for (int a = 0; a < 9; ++a) y[a] += __shfl_xor(y[a], off, 32);
  }

  if (v < 9) {                               // fold CART_B at store time
    float o = 0.f;
#pragma unroll
    for (int a = 0; a < 9; ++a) o += y[a] * c_BM[a * 9 + v];
    out[(size_t)e * 9 + v] = o;
  }
}

// ---- host launcher ---------------------------------------------------------
extern "C" void kernel_launch(void* const* d_in, const int* in_sizes, int n_in,
                              void* d_out, int out_size, void* d_ws, size_t ws_size,
                              hipStream_t stream) {
  const float* nf     = (const float*)d_in[0];
  const int*   layout = (const int*)d_in[1];
  const float* w_ss   = (const float*)d_in[2];
  const float* w_vv0  = (const float*)d_in[3];
  const float* w_sv   = (const float*)d_in[4];
  const float* w_vs   = (const float*)d_in[5];
  const float* w_vv1  = (const float*)d_in[6];
  const float* w_vv2  = (const float*)d_in[7];
  float* out = (float*)d_out;

  const int n_nodes = in_sizes[0] / 128;
  const int n_pairs = in_sizes[1] / 2;

  // workspace: 448 floats per node (prescaled transforms)
  float* ws   = (float*)d_ws;
  float* p_ss = ws;
  float* p_sv = p_ss + (size_t)n_nodes * 32;
  float* t0   = p_sv + (size_t)n_nodes * 32;
  float* ts   = t0   + (size_t)n_nodes * 96;
  float* t1   = ts   + (size_t)n_nodes * 96;
  float* t2   = t1   + (size_t)n_nodes * 96;

  const int rt_s = (n_nodes + 15) / 16;
  const int rt_v = (3 * n_nodes + 15) / 16;
  const int tiles = 4 * rt_s + 8 * rt_v;

  // Kernel A: 4 waves/block, one wave per 16x16 WMMA tile
  node_gemm_wmma<<<(tiles + 3) / 4, 128, 0, stream>>>(
      nf, w_ss, w_sv, w_vv0, w_vs, w_vv1, w_vv2,
      p_ss, p_sv, t0, ts, t1, t2, n_nodes, rt_s, rt_v);

  // Kernel B: 8 waves/block, one wave per pair
  pair_contract<<<(n_pairs + 7) / 8, 256, 0, stream>>>(
      nf, layout, p_ss, p_sv, t0, ts, t1, t2, out, n_pairs);
}